// Encoder_modified_1176821039648
// MI455X (gfx1250) — compile-verified
//
#include <hip/hip_runtime.h>

// ---------------------------------------------------------------------------
// GraphSAGE encoder for MI455X (gfx1250, wave32, WMMA + TDM).
//   agg = scatter-mean over edges; h = agg@W_l + b_l + x@W_r  (fp32 WMMA)
//   out1 = 10*norm(h)@norm(W1), out2 = 10*norm(h)@norm(W2)    (fused WMMA)
// Roofline: scatter (204.8M f32 atomics, L2-resident) dominates; the GEMMs
// are small (6.55 GFLOP) -> fp32 V_WMMA_F32_16X16X4_F32 for output fidelity.
// Weights are staged Global->LDS once per block via the Tensor Data Mover
// (TENSOR_LOAD_TO_LDS + s_wait_tensorcnt), so B fragments come from LDS and
// the vmem pipe is reserved for the streamed A operands (agg / x rows).
// ---------------------------------------------------------------------------

typedef __attribute__((ext_vector_type(2))) float v2f;
typedef __attribute__((ext_vector_type(8))) float v8f;
typedef __attribute__((ext_vector_type(4))) unsigned int v4u;
typedef __attribute__((ext_vector_type(4))) int v4i;
typedef __attribute__((ext_vector_type(8))) int v8i;

#define DIM 128

#if __has_builtin(__builtin_amdgcn_tensor_load_to_lds)
#define HAVE_TDM 1
#else
#define HAVE_TDM 0
#endif

#if HAVE_TDM
// Issue one TDM 2-D tile load: gptr[rows][cols] (f32, row-major, contiguous)
// into LDS at byte offset lds_off. tile == tensor. Tracked by TENSORcnt.
__device__ __forceinline__ void tdm_load_2d(unsigned int lds_off,
                                            const void* gptr,
                                            unsigned int rows,   // dim1
                                            unsigned int cols) { // dim0 (contiguous)
  const unsigned long long ga = (unsigned long long)(uintptr_t)gptr;
  v4u g0;
  g0.x = 1u;                                   // count=1, user descriptor
  g0.y = lds_off;                              // lds_addr [63:32]
  g0.z = (unsigned int)(ga & 0xffffffffu);     // global_addr [95:64]
  g0.w = (unsigned int)((ga >> 32) & 0x01ffffffu) | (2u << 30); // addr hi | type=2
  v8i g1;
  g1[0] = 0x00020000;                          // wg_mask=0, data_size=2 (4B)
  g1[1] = (int)(cols << 16);                   // tensor_dim0 lo16 at [63:48]
  g1[2] = (int)((cols >> 16) | (rows << 16));  // dim0 hi | tensor_dim1 lo
  g1[3] = (int)((rows >> 16) | (cols << 16));  // dim1 hi | tile_dim0
  g1[4] = (int)rows;                           // tile_dim1 (tile_dim2 = 0)
  g1[5] = (int)cols;                           // tensor_dim0_stride lo32
  g1[6] = 0;
  g1[7] = 0;                                   // dim1_stride unused for 2-D
  const v4i z4 = {0, 0, 0, 0};
#if defined(__clang_major__) && __clang_major__ >= 23
  const v8i z8 = {0, 0, 0, 0, 0, 0, 0, 0};
  __builtin_amdgcn_tensor_load_to_lds(g0, g1, z4, z4, z8, 0);
#else
  __builtin_amdgcn_tensor_load_to_lds(g0, g1, z4, z4, 0);
#endif
}
#endif

// ---------------- zero workspace ----------------
__global__ __launch_bounds__(256) void k_zero(float* __restrict__ p, long n) {
  long i = (long)blockIdx.x * blockDim.x + threadIdx.x;
  long stride = (long)gridDim.x * blockDim.x;
  for (; i < n; i += stride) p[i] = 0.0f;
}

// ---------------- edge scatter: agg[dst] += x[src]; deg[dst] += 1 ----------
__global__ __launch_bounds__(256) void k_scatter(const float* __restrict__ x,
                                                 const long long* __restrict__ src,
                                                 const long long* __restrict__ dst,
                                                 float* __restrict__ agg,
                                                 float* __restrict__ deg,
                                                 int E) {
  const int e    = (int)((blockIdx.x * blockDim.x + threadIdx.x) >> 5);
  const int lane = threadIdx.x & 31;
  if (e >= E) return;
  const int s = (int)src[e];
  const int d = (int)dst[e];
  const float4 v = *(const float4*)(x + (long)s * DIM + lane * 4);
  float* a = agg + (long)d * DIM + lane * 4;
  __hip_atomic_fetch_add(a + 0, v.x, __ATOMIC_RELAXED, __HIP_MEMORY_SCOPE_AGENT);
  __hip_atomic_fetch_add(a + 1, v.y, __ATOMIC_RELAXED, __HIP_MEMORY_SCOPE_AGENT);
  __hip_atomic_fetch_add(a + 2, v.z, __ATOMIC_RELAXED, __HIP_MEMORY_SCOPE_AGENT);
  __hip_atomic_fetch_add(a + 3, v.w, __ATOMIC_RELAXED, __HIP_MEMORY_SCOPE_AGENT);
  if (lane == 0)
    __hip_atomic_fetch_add(deg + d, 1.0f, __ATOMIC_RELAXED, __HIP_MEMORY_SCOPE_AGENT);
}

// ---------------- h = (agg/deg)@W_l + b_l + x@W_r  (fp32 WMMA) -------------
// Block: 256 threads = 8 waves covering 16 rows x 128 cols; wave w covers
// columns [16w, 16w+16). Dynamic LDS: W_l tile (64KB) + W_r tile (64KB),
// filled by TDM; B fragments then come from LDS.
__global__ __launch_bounds__(256) void k_gemm(const float* __restrict__ agg,
                                              const float* __restrict__ deg,
                                              const float* __restrict__ x,
                                              const float* __restrict__ Wl,
                                              const float* __restrict__ bl,
                                              const float* __restrict__ Wr,
                                              float* __restrict__ h,
                                              int N) {
  extern __shared__ float lds[];
  float* lWl = lds;
  float* lWr = lds + DIM * DIM;

#if HAVE_TDM
  if (threadIdx.x < 32) {                    // wave 0 issues both TDM tiles
    tdm_load_2d((unsigned int)(uintptr_t)lWl, Wl, DIM, DIM);
    tdm_load_2d((unsigned int)(uintptr_t)lWr, Wr, DIM, DIM);
    __builtin_amdgcn_s_wait_tensorcnt(0);
  }
#else
  for (int i = threadIdx.x; i < DIM * DIM; i += 256) {
    lWl[i] = Wl[i];
    lWr[i] = Wr[i];
  }
#endif
  __syncthreads();

  const int wave = threadIdx.x >> 5;      // 0..7
  const int lane = threadIdx.x & 31;
  const int half = lane >> 4;             // K pair (A/B) and M half (C/D)
  const int l15  = lane & 15;
  const int row0 = blockIdx.x * 16;

  int rowA = row0 + l15;                  // A-fragment row (M = lane&15)
  if (rowA >= N) rowA = N - 1;            // clamp: keep EXEC all-1 for WMMA
  const float rdeg = 1.0f / fmaxf(deg[rowA], 1.0f);
  const float* aggRow = agg + (long)rowA * DIM;
  const float* xRow   = x   + (long)rowA * DIM;
  const int col = wave * 16 + l15;        // B-fragment column (N = lane&15)

  v8f acc = {};
  for (int k0 = 0; k0 < DIM; k0 += 4) {
    const int kb = k0 + half * 2;
    v2f aL, aR, bL, bR;
    const float2 a0 = *(const float2*)(aggRow + kb);
    const float2 a1 = *(const float2*)(xRow + kb);
    aL.x = a0.x * rdeg;  aL.y = a0.y * rdeg;
    aR.x = a1.x;         aR.y = a1.y;
    bL.x = lWl[kb * DIM + col];
    bL.y = lWl[(kb + 1) * DIM + col];
    bR.x = lWr[kb * DIM + col];
    bR.y = lWr[(kb + 1) * DIM + col];
    acc = __builtin_amdgcn_wmma_f32_16x16x4_f32(false, aL, false, bL,
                                                (short)0, acc, false, false);
    acc = __builtin_amdgcn_wmma_f32_16x16x4_f32(false, aR, false, bR,
                                                (short)0, acc, false, false);
  }

  const float bias = bl[col];
  float* hc = h + (long)(row0 + half * 8) * DIM + col;
  if (row0 + 16 <= N) {                   // full tile: branch-free store clause
#pragma unroll
    for (int i = 0; i < 8; ++i) hc[(long)i * DIM] = acc[i] + bias;
  } else {
#pragma unroll
    for (int i = 0; i < 8; ++i) {
      const int row = row0 + i + half * 8;
      if (row < N) h[(long)row * DIM + col] = acc[i] + bias;
    }
  }
}

// ---------------- per-row scale: rinv = 10 / max(||h_row||, eps) -----------
__global__ __launch_bounds__(256) void k_rownorm(const float* __restrict__ h,
                                                 float* __restrict__ rinv,
                                                 int N) {
  const int row  = (int)((blockIdx.x * blockDim.x + threadIdx.x) >> 5);
  const int lane = threadIdx.x & 31;
  if (row >= N) return;
  const float4 v = *(const float4*)(h + (long)row * DIM + lane * 4);
  float s = v.x * v.x + v.y * v.y + v.z * v.z + v.w * v.w;
#pragma unroll
  for (int o = 16; o > 0; o >>= 1) s += __shfl_xor(s, o, 32);
  if (lane == 0) rinv[row] = 10.0f / fmaxf(sqrtf(s), 1e-12f);
}

// ---------------- normalize head weights into padded [128 x 32] ------------
// cols 0..9  <- W1 (128x10), cols 10..29 <- W2 (128x20), cols 30..31 zero.
__global__ __launch_bounds__(32) void k_normw(const float* __restrict__ W1,
                                              const float* __restrict__ W2,
                                              float* __restrict__ Wn) {
  const int c = threadIdx.x;  // 0..31
  float ss = 0.0f;
  for (int k = 0; k < DIM; ++k) {
    float w = (c < 10) ? W1[k * 10 + c] : ((c < 30) ? W2[k * 20 + (c - 10)] : 0.0f);
    ss += w * w;
  }
  const float inv = (c < 30) ? (1.0f / fmaxf(sqrtf(ss), 1e-12f)) : 0.0f;
  for (int k = 0; k < DIM; ++k) {
    float w = (c < 10) ? W1[k * 10 + c] : ((c < 30) ? W2[k * 20 + (c - 10)] : 0.0f);
    Wn[k * 32 + c] = w * inv;
  }
}

// ---------------- heads: [16 rows] x [32 cols] WMMA tile per block ---------
// Block: 64 threads = 2 waves; wave w covers cols [16w, 16w+16).
// Wn tile (16KB) staged into LDS by TDM.
__global__ __launch_bounds__(64) void k_heads(const float* __restrict__ h,
                                              const float* __restrict__ rinv,
                                              const float* __restrict__ Wn,
                                              float* __restrict__ out1,
                                              float* __restrict__ out2,
                                              int N) {
  __shared__ float lWn[DIM * 32];

#if HAVE_TDM
  if (threadIdx.x < 32) {                 // wave 0 issues the TDM tile
    tdm_load_2d((unsigned int)(uintptr_t)lWn, Wn, DIM, 32);
    __builtin_amdgcn_s_wait_tensorcnt(0);
  }
#else
  for (int i = threadIdx.x; i < DIM * 32; i += 64) lWn[i] = Wn[i];
#endif
  __syncthreads();

  const int wave = threadIdx.x >> 5;      // 0..1
  const int lane = threadIdx.x & 31;
  const int half = lane >> 4;
  const int l15  = lane & 15;
  const int row0 = blockIdx.x * 16;

  int rowA = row0 + l15;
  if (rowA >= N) rowA = N - 1;
  const float sc = rinv[rowA];            // folds 10/||h|| into A fragment
  const float* hRow = h + (long)rowA * DIM;
  const int col = wave * 16 + l15;

  v8f acc = {};
  for (int k0 = 0; k0 < DIM; k0 += 4) {
    const int kb = k0 + half * 2;
    v2f a, b;
    const float2 a0 = *(const float2*)(hRow + kb);
    a.x = a0.x * sc;
    a.y = a0.y * sc;
    b.x = lWn[kb * 32 + col];
    b.y = lWn[(kb + 1) * 32 + col];
    acc = __builtin_amdgcn_wmma_f32_16x16x4_f32(false, a, false, b,
                                                (short)0, acc, false, false);
  }

#pragma unroll
  for (int i = 0; i < 8; ++i) {
    const int row = row0 + i + half * 8;
    if (row < N) {
      if (col < 10)      out1[(long)row * 10 + col]        = acc[i];
      else if (col < 30) out2[(long)row * 20 + (col - 10)] = acc[i];
    }
  }
}

// ---------------------------------------------------------------------------
extern "C" void kernel_launch(void* const* d_in, const int* in_sizes, int n_in,
                              void* d_out, int out_size, void* d_ws, size_t ws_size,
                              hipStream_t stream) {
  (void)n_in; (void)out_size; (void)ws_size;
  const float*     x  = (const float*)d_in[0];
  const long long* ei = (const long long*)d_in[1];   // int64 [2, E]
  const float*     Wl = (const float*)d_in[2];
  const float*     bl = (const float*)d_in[3];
  const float*     Wr = (const float*)d_in[4];
  const float*     W1 = (const float*)d_in[5];
  const float*     W2 = (const float*)d_in[6];

  const int N = in_sizes[0] / DIM;
  const int E = in_sizes[1] / 2;

  // Output layout: out1 [N,10] | out2 [N,20] | h [N,128]
  float* out1 = (float*)d_out;
  float* out2 = out1 + (long)N * 10;
  float* h    = out1 + (long)N * 30;

  // Workspace layout: agg [N,128] | deg [N] | rinv [N] | Wn [128,32]
  float* agg  = (float*)d_ws;
  float* deg  = agg  + (long)N * DIM;
  float* rinv = deg  + N;
  float* Wn   = rinv + N;

  const long nzero = (long)N * DIM + N;   // agg + deg are contiguous
  k_zero<<<1024, 256, 0, stream>>>(agg, nzero);

  k_scatter<<<(E + 7) / 8, 256, 0, stream>>>(x, ei, ei + E, agg, deg, E);

  // 128KB dynamic LDS: W_l + W_r tiles staged by TDM
  k_gemm<<<(N + 15) / 16, 256, 2 * DIM * DIM * (int)sizeof(float), stream>>>(
      agg, deg, x, Wl, bl, Wr, h, N);

  k_rownorm<<<(N + 7) / 8, 256, 0, stream>>>(h, rinv, N);

  k_normw<<<1, 32, 0, stream>>>(W1, W2, Wn);

  k_heads<<<(N + 15) / 16, 64, 0, stream>>>(h, rinv, Wn, out1, out2, N);
}